// GraphAttentionLayer_22179211116913
// MI455X (gfx1250) — compile-verified
//
#include <hip/hip_runtime.h>
#include <hip/hip_bf16.h>

// ---------------------------------------------------------------------------
// GAT layer on MI455X (gfx1250, wave32).
//   h     = x @ W                         (WMMA f32 16x16x4, memory-bound)
//   a_s/d = per-node per-head logits
//   edge passes: leakyrelu -> seg-max -> exp/seg-sum -> weighted scatter
// ---------------------------------------------------------------------------

#define NUM_HEADS 4
#define HEAD_DIM 32
#define F_IN 256
#define F_OUT 128            // NUM_HEADS * HEAD_DIM
#define LDA 260              // padded LDS stride (floats): 16 rows hit 64 distinct banks

typedef __attribute__((ext_vector_type(2))) float v2f;
typedef __attribute__((ext_vector_type(8))) float v8f;

// ---------------------------------------------------------------- init ------
__global__ __launch_bounds__(256) void gat_init(float* __restrict__ out,
                                                float* __restrict__ emax,
                                                float* __restrict__ asum,
                                                int N) {
    int tid = blockIdx.x * blockDim.x + threadIdx.x;
    if (tid < N * F_OUT) out[tid] = 0.0f;
    if (tid < N * NUM_HEADS) { emax[tid] = -3.0e38f; asum[tid] = 0.0f; }
}

// ------------------------------------------------------- WMMA f32 GEMM ------
// One block = 8 waves = one 16-row tile of h. Wave w computes columns
// [16w, 16w+16). A tile staged in LDS (padded), B streamed from global (W is
// 128 KB -> L2/L1 resident).
__global__ __launch_bounds__(256) void gat_gemm(const float* __restrict__ x,
                                                const float* __restrict__ W,
                                                float* __restrict__ h) {
    __shared__ float As[16 * LDA];

    const int t = threadIdx.x;
    const int rowBase = blockIdx.x * 16;

    // Cooperative load of the 16x256 A tile (1024 float4, 4 per thread).
    #pragma unroll
    for (int j = 0; j < 4; ++j) {
        int f = t + 256 * j;          // float4 index 0..1023
        int r = f >> 6;               // 64 float4 per row
        int c = (f & 63) << 2;
        float4 v = *(const float4*)(x + (size_t)(rowBase + r) * F_IN + c);
        *(float4*)(&As[r * LDA + c]) = v;
    }
    __syncthreads();

    const int wave = t >> 5;
    const int lane = t & 31;
    const int m    = lane & 15;               // M index for A / N index for B
    const int koff = (lane >> 4) << 1;        // 0 or 2 (K pair select)
    const int col  = (wave << 4) + m;         // output column 0..127

    const float* Wp = W + col;
    v8f acc = {};

    #pragma unroll 8
    for (int k = 0; k < F_IN; k += 4) {
        // A: lane m holds (K=k+koff, k+koff+1) of row m  -> ds_load_b64
        v2f a = *(const v2f*)(&As[m * LDA + k + koff]);
        // B: lane m holds rows (k+koff, k+koff+1), column `col`
        v2f b;
        b.x = Wp[(size_t)(k + koff) * F_OUT];
        b.y = Wp[(size_t)(k + koff + 1) * F_OUT];
        acc = __builtin_amdgcn_wmma_f32_16x16x4_f32(
            /*neg_a=*/false, a, /*neg_b=*/false, b,
            /*c_mod=*/(short)0, acc, /*reuse_a=*/false, /*reuse_b=*/false);
    }

    // C/D layout: VGPR v holds row (v + 8*(lane>=16)), column lane&15.
    const int roff = (lane >> 4) << 3;
    float* hp = h + (size_t)rowBase * F_OUT + col;
    #pragma unroll
    for (int v = 0; v < 8; ++v)
        hp[(size_t)(v + roff) * F_OUT] = acc[v];
}

// Scalar tail for N not divisible by 16 (not launched when N%16==0).
__global__ __launch_bounds__(256) void gat_gemm_tail(const float* __restrict__ x,
                                                     const float* __restrict__ W,
                                                     float* __restrict__ h,
                                                     int N, int rowStart) {
    int idx = blockIdx.x * blockDim.x + threadIdx.x;
    int total = (N - rowStart) * F_OUT;
    if (idx >= total) return;
    int r = rowStart + idx / F_OUT;
    int c = idx % F_OUT;
    float s = 0.0f;
    for (int k = 0; k < F_IN; ++k)
        s += x[(size_t)r * F_IN + k] * W[(size_t)k * F_OUT + c];
    h[(size_t)r * F_OUT + c] = s;
}

// ----------------------------------------------- per-node attention logits --
__global__ __launch_bounds__(256) void gat_alpha(const float* __restrict__ h,
                                                 const float* __restrict__ a_src,
                                                 const float* __restrict__ a_dst,
                                                 float* __restrict__ asrc,
                                                 float* __restrict__ adst,
                                                 int N) {
    int tid = blockIdx.x * blockDim.x + threadIdx.x;
    if (tid >= N * NUM_HEADS) return;
    int n  = tid >> 2;
    int hd = tid & 3;
    const float* hp = h + (size_t)n * F_OUT + hd * HEAD_DIM;
    const float* as = a_src + hd * HEAD_DIM;
    const float* ad = a_dst + hd * HEAD_DIM;
    float s1 = 0.0f, s2 = 0.0f;
    #pragma unroll
    for (int i = 0; i < HEAD_DIM; i += 4) {
        float4 hv = *(const float4*)(hp + i);
        float4 av = *(const float4*)(as + i);
        float4 dv = *(const float4*)(ad + i);
        s1 += hv.x * av.x + hv.y * av.y + hv.z * av.z + hv.w * av.w;
        s2 += hv.x * dv.x + hv.y * dv.y + hv.z * dv.z + hv.w * dv.w;
    }
    asrc[tid] = s1;
    adst[tid] = s2;
}

__device__ __forceinline__ float leaky(float v) { return v > 0.0f ? v : 0.2f * v; }

// ------------------------------------------------- edge pass 1: segment max -
__global__ __launch_bounds__(256) void gat_edge_max(const int* __restrict__ src,
                                                    const int* __restrict__ dst,
                                                    const float* __restrict__ asrc,
                                                    const float* __restrict__ adst,
                                                    float* __restrict__ emax,
                                                    int E) {
    int e = blockIdx.x * blockDim.x + threadIdx.x;
    if (e >= E) return;
    int s = src[e], d = dst[e];
    #pragma unroll
    for (int hd = 0; hd < NUM_HEADS; ++hd) {
        float v = leaky(asrc[s * NUM_HEADS + hd] + adst[d * NUM_HEADS + hd]);
        __hip_atomic_fetch_max(&emax[d * NUM_HEADS + hd], v,
                               __ATOMIC_RELAXED, __HIP_MEMORY_SCOPE_AGENT);
    }
}

// ------------------------------------------------- edge pass 2: segment sum -
__global__ __launch_bounds__(256) void gat_edge_sum(const int* __restrict__ src,
                                                    const int* __restrict__ dst,
                                                    const float* __restrict__ asrc,
                                                    const float* __restrict__ adst,
                                                    const float* __restrict__ emax,
                                                    float* __restrict__ asum,
                                                    int E) {
    int e = blockIdx.x * blockDim.x + threadIdx.x;
    if (e >= E) return;
    int s = src[e], d = dst[e];
    #pragma unroll
    for (int hd = 0; hd < NUM_HEADS; ++hd) {
        float v = leaky(asrc[s * NUM_HEADS + hd] + adst[d * NUM_HEADS + hd]);
        float w = __expf(v - emax[d * NUM_HEADS + hd]);
        atomicAdd(&asum[d * NUM_HEADS + hd], w);
    }
}

// -------------------------------------- edge pass 3: weighted scatter-add ---
// One wave per edge; lane l handles 4 consecutive features (float4 gather of
// h[src], 4 global_atomic_add_f32 into out[dst], which is L2-resident).
__global__ __launch_bounds__(256) void gat_aggregate(const int* __restrict__ src,
                                                     const int* __restrict__ dst,
                                                     const float* __restrict__ asrc,
                                                     const float* __restrict__ adst,
                                                     const float* __restrict__ emax,
                                                     const float* __restrict__ asum,
                                                     const float* __restrict__ h,
                                                     float* __restrict__ out,
                                                     int E) {
    int e = blockIdx.x * 8 + (threadIdx.x >> 5);
    if (e >= E) return;
    int lane = threadIdx.x & 31;
    int s = src[e], d = dst[e];
    int elem = lane << 2;          // 0..124
    int hd   = elem >> 5;          // head index

    float v = leaky(asrc[s * NUM_HEADS + hd] + adst[d * NUM_HEADS + hd]);
    float w = __expf(v - emax[d * NUM_HEADS + hd]) /
              (asum[d * NUM_HEADS + hd] + 1e-10f);

    float4 hv = *(const float4*)(h + (size_t)s * F_OUT + elem);
    float* op = out + (size_t)d * F_OUT + elem;
    atomicAdd(op + 0, w * hv.x);
    atomicAdd(op + 1, w * hv.y);
    atomicAdd(op + 2, w * hv.z);
    atomicAdd(op + 3, w * hv.w);
}

// ---------------------------------------------------------------------------
extern "C" void kernel_launch(void* const* d_in, const int* in_sizes, int n_in,
                              void* d_out, int out_size, void* d_ws, size_t ws_size,
                              hipStream_t stream) {
    const float* x     = (const float*)d_in[0];
    const int*   ei    = (const int*)d_in[1];
    const float* W     = (const float*)d_in[2];
    const float* a_src = (const float*)d_in[3];
    const float* a_dst = (const float*)d_in[4];
    float*       out   = (float*)d_out;

    const int N = in_sizes[0] / F_IN;
    const int E = in_sizes[1] / 2;
    const int* src = ei;
    const int* dst = ei + E;

    // Workspace carve-up: h (N*128) | asrc (N*4) | adst (N*4) | emax (N*4) | asum (N*4)
    float* h    = (float*)d_ws;
    float* asrc = h    + (size_t)N * F_OUT;
    float* adst = asrc + (size_t)N * NUM_HEADS;
    float* emax = adst + (size_t)N * NUM_HEADS;
    float* asum = emax + (size_t)N * NUM_HEADS;

    // 0) init accumulators + output
    {
        int total = N * F_OUT;
        gat_init<<<(total + 255) / 256, 256, 0, stream>>>(out, emax, asum, N);
    }

    // 1) h = x @ W via WMMA (full 16-row tiles), scalar tail if needed
    {
        int fullTiles = N / 16;
        if (fullTiles > 0)
            gat_gemm<<<fullTiles, 256, 0, stream>>>(x, W, h);
        int rowStart = fullTiles * 16;
        int rem = N - rowStart;
        if (rem > 0) {
            int total = rem * F_OUT;
            gat_gemm_tail<<<(total + 255) / 256, 256, 0, stream>>>(x, W, h, N, rowStart);
        }
    }

    // 2) per-node logits
    {
        int total = N * NUM_HEADS;
        gat_alpha<<<(total + 255) / 256, 256, 0, stream>>>(h, a_src, a_dst, asrc, adst, N);
    }

    // 3) segment max over incoming edges
    gat_edge_max<<<(E + 255) / 256, 256, 0, stream>>>(src, dst, asrc, adst, emax, E);

    // 4) segment sum of exp(e - max)
    gat_edge_sum<<<(E + 255) / 256, 256, 0, stream>>>(src, dst, asrc, adst, emax, asum, E);

    // 5) normalized weighted scatter-add (wave per edge)
    gat_aggregate<<<(E + 7) / 8, 256, 0, stream>>>(src, dst, asrc, adst, emax, asum, h, out, E);
}